// SA_content_5454608466075
// MI455X (gfx1250) — compile-verified
//
#include <hip/hip_runtime.h>
#include <math.h>

// MI455X / gfx1250 (wave32). Fused flash-attention formulation of the
// reference: Q=(w1 x)^T, K=(w2 x)^T, V=(w3 x)^T, out = softmax(Q K^T) V + x.
// Never materializes the 256 MiB attention matrix; Q/K/V (12 MiB) are
// L2-resident (192 MB L2). fp32 WMMA (V_WMMA_F32_16X16X4_F32) keeps numerics
// at reference precision; FLOP count (~17 GFLOP) is tiny, so the win is in
// data movement: async global->LDS DMA (ASYNCcnt) double-buffering + padded
// LDS tiles + batched ds_load->wmma clauses instead of load/wait/wmma chains.

typedef float v2f __attribute__((ext_vector_type(2)));
typedef float v8f __attribute__((ext_vector_type(8)));

constexpr int NB = 4;      // batch
constexpr int CH = 64;     // channels (= head dim)
constexpr int NN = 4096;   // H*W (= sequence length)

constexpr int KS = 68;     // K-tile LDS stride: b64 reads conflict-free, rows 16B-aligned
constexpr int VS = 72;     // V-tile LDS stride: both lane-halves' b32 reads hit disjoint banks
constexpr int PS = 36;     // P-tile LDS stride: b64 reads conflict-free incl. +2 half shift
constexpr int WS = 68;     // weight LDS stride (same pattern as K)

// gfx1250 async global->LDS copy (128b). Per-lane: copies 16 bytes from global gaddr to the LDS address in ldsaddr.
// Tracked by ASYNCcnt; waited with s_wait_asynccnt.
__device__ __forceinline__ void async_copy_b128(const float* gaddr, const float* ldsaddr) {
    asm volatile("global_load_async_to_lds_b128 %0, %1, off"
                 :: "v"((unsigned int)(unsigned long long)ldsaddr), "v"(gaddr)
                 : "memory");
}
__device__ __forceinline__ void wait_async0() {
    asm volatile("s_wait_asynccnt 0x0" ::: "memory");
}

// ---------------------------------------------------------------------------
// Kernel 1: Q/K/V = w{1,2,3} @ x + b, stored [B][N][64] row-major.
// Weights staged once per block into LDS via async DMA, shared by 8 waves.
// ---------------------------------------------------------------------------
__global__ __launch_bounds__(256) void qkv_kernel(
    const float* __restrict__ x,
    const float* __restrict__ w1, const float* __restrict__ b1,
    const float* __restrict__ w2, const float* __restrict__ b2,
    const float* __restrict__ w3, const float* __restrict__ b3,
    float* __restrict__ Q, float* __restrict__ K, float* __restrict__ V)
{
    __shared__ alignas(16) float sW[3 * 64 * WS];   // padded weight tiles
    __shared__ alignas(16) float sBias[3 * 64];

    const int tid  = threadIdx.x;
    const int lane = tid & 31;
    const int wv   = tid >> 5;                 // 0..7
    const int gw   = blockIdx.x * 8 + wv;      // 0..1023
    const int b    = gw >> 8;                  // batch
    const int nbase = (gw & 255) * 16;
    const int lo = lane & 15, hi = lane >> 4;

    // --- stage all three weight matrices (48 KB) into LDS asynchronously ---
    const float* Wg[3] = { w1, w2, w3 };
    #pragma unroll
    for (int i = 0; i < 12; ++i) {             // 3072 float4 slots / 256 threads
        const int s  = tid + 256 * i;
        const int m  = s >> 10;                // matrix
        const int r  = s & 1023;
        const int o  = r >> 4;
        const int c4 = (r & 15) << 2;
        async_copy_b128(Wg[m] + o * CH + c4, sW + (m * 64 + o) * WS + c4);
    }
    if (tid < 192) {                           // biases (tiny, synchronous)
        const float* Bg[3] = { b1, b2, b3 };
        sBias[tid] = Bg[tid >> 6][tid & 63];
    }

    // A operands (x^T tile [16 n x 64 c]) while DMA is in flight.
    v2f xa[16];
    const float* xb = x + (size_t)b * CH * NN;
    #pragma unroll
    for (int k = 0; k < 16; ++k) {
        const int c = 4 * k + 2 * hi;
        xa[k].x = xb[(size_t)c * NN + nbase + lo];
        xa[k].y = xb[(size_t)(c + 1) * NN + nbase + lo];
    }

    wait_async0();
    __syncthreads();

    float* Os[3] = { Q, K, V };
    #pragma unroll
    for (int m = 0; m < 3; ++m) {
        const float* wl = sW + m * 64 * WS;
        float*       o  = Os[m] + (size_t)b * NN * CH;
        #pragma unroll
        for (int ob = 0; ob < 4; ++ob) {
            const int obase = ob * 16;
            // batch the 16 B-operand ds_loads, then a clean WMMA burst
            v2f wb[16];
            #pragma unroll
            for (int k = 0; k < 16; ++k)
                wb[k] = *(const v2f*)&wl[(obase + lo) * WS + 4 * k + 2 * hi];
            v8f acc = {};
            #pragma unroll
            for (int k = 0; k < 16; ++k)
                acc = __builtin_amdgcn_wmma_f32_16x16x4_f32(
                        false, xa[k], false, wb[k], (short)0, acc, false, false);
            const float bv = sBias[m * 64 + obase + lo];
            #pragma unroll
            for (int v = 0; v < 8; ++v) {
                const int row = nbase + v + 8 * hi;
                o[(size_t)row * CH + obase + lo] = acc[v] + bv;
            }
        }
    }
}

// ---------------------------------------------------------------------------
// Kernel 2: fused softmax(Q K^T) V + residual. 4 waves/block, each wave owns
// 16 query rows; K/V 32-key tiles double-buffered in LDS via async DMA.
// ---------------------------------------------------------------------------
__global__ __launch_bounds__(128) void attn_kernel(
    const float* __restrict__ Q, const float* __restrict__ K,
    const float* __restrict__ V, const float* __restrict__ x,
    float* __restrict__ out)
{
    __shared__ alignas(16) float sK[2][32 * KS];
    __shared__ alignas(16) float sV[2][32 * VS];
    __shared__ alignas(16) float sP[4 * 16 * PS];

    const int tid  = threadIdx.x;
    const int lane = tid & 31;
    const int wv   = tid >> 5;
    const int lo = lane & 15, hi = lane >> 4;
    const int b     = blockIdx.x >> 6;
    const int nbase = (blockIdx.x & 63) * 64 + wv * 16;

    const float* Qb = Q + (size_t)b * NN * CH;
    const float* Kb = K + (size_t)b * NN * CH;
    const float* Vb = V + (size_t)b * NN * CH;

    // Register-resident Q A-operands.
    v2f qa[16];
    #pragma unroll
    for (int k = 0; k < 16; ++k) {
        const int c = 4 * k + 2 * hi;
        qa[k].x = Qb[(size_t)(nbase + lo) * CH + c];
        qa[k].y = Qb[(size_t)(nbase + lo) * CH + c + 1];
    }

    v8f acc[4] = { v8f{}, v8f{}, v8f{}, v8f{} };
    float mrow[8], lrow[8];
    #pragma unroll
    for (int v = 0; v < 8; ++v) { mrow[v] = -3.0e38f; lrow[v] = 0.0f; }

    float* myP = sP + wv * 16 * PS;

    // async-stage one 32-key K/V tile into LDS buffer `buf`
    auto stage = [&](int mbase, int buf) {
        #pragma unroll
        for (int i = 0; i < 4; ++i) {
            const int s  = tid + 128 * i;      // 512 float4 slots
            const int mm = s >> 4;
            const int c4 = (s & 15) << 2;
            async_copy_b128(Kb + (size_t)(mbase + mm) * CH + c4, &sK[buf][mm * KS + c4]);
            async_copy_b128(Vb + (size_t)(mbase + mm) * CH + c4, &sV[buf][mm * VS + c4]);
        }
    };

    constexpr int NITER = NN / 32;
    stage(0, 0);
    int cur = 0;

    for (int it = 0; it < NITER; ++it) {
        wait_async0();
        __syncthreads();                        // tile[cur] ready; prev compute done
        if (it + 1 < NITER) stage((it + 1) * 32, cur ^ 1);   // DMA overlaps compute

        const float* cK = sK[cur];
        const float* cV = sV[cur];

        // ---- S = Q K^T : two 16x16 tiles, batched ds_load -> WMMA bursts ----
        v8f s0 = {}, s1 = {};
        #pragma unroll
        for (int half = 0; half < 2; ++half) {
            v2f kb0[8], kb1[8];
            #pragma unroll
            for (int k = 0; k < 8; ++k) {
                const int c = 4 * (half * 8 + k) + 2 * hi;
                kb0[k] = *(const v2f*)&cK[lo * KS + c];
                kb1[k] = *(const v2f*)&cK[(16 + lo) * KS + c];
            }
            #pragma unroll
            for (int k = 0; k < 8; ++k) {
                s0 = __builtin_amdgcn_wmma_f32_16x16x4_f32(
                        false, qa[half * 8 + k], false, kb0[k], (short)0, s0, false, false);
                s1 = __builtin_amdgcn_wmma_f32_16x16x4_f32(
                        false, qa[half * 8 + k], false, kb1[k], (short)0, s1, false, false);
            }
        }

        // ---- Online softmax over this 32-key slab ----
        float alpha[8];
        #pragma unroll
        for (int v = 0; v < 8; ++v) {
            float mx = fmaxf(s0[v], s1[v]);
            mx = fmaxf(mx, __shfl_xor(mx, 1, 32));
            mx = fmaxf(mx, __shfl_xor(mx, 2, 32));
            mx = fmaxf(mx, __shfl_xor(mx, 4, 32));
            mx = fmaxf(mx, __shfl_xor(mx, 8, 32));
            const float mn = fmaxf(mrow[v], mx);
            alpha[v] = __expf(mrow[v] - mn);
            const float p0 = __expf(s0[v] - mn);
            const float p1 = __expf(s1[v] - mn);
            float rs = p0 + p1;
            rs += __shfl_xor(rs, 1, 32);
            rs += __shfl_xor(rs, 2, 32);
            rs += __shfl_xor(rs, 4, 32);
            rs += __shfl_xor(rs, 8, 32);
            lrow[v] = lrow[v] * alpha[v] + rs;
            mrow[v] = mn;
            myP[(v + 8 * hi) * PS + lo]      = p0;   // C/D layout -> LDS
            myP[(v + 8 * hi) * PS + 16 + lo] = p1;
        }
        #pragma unroll
        for (int d = 0; d < 4; ++d)
            #pragma unroll
            for (int v = 0; v < 8; ++v) acc[d][v] *= alpha[v];

        __syncthreads();                        // order P stores before A-layout loads

        // ---- acc += P V ----
        v2f pa[8];
        #pragma unroll
        for (int m0 = 0; m0 < 8; ++m0)
            pa[m0] = *(const v2f*)&myP[lo * PS + 4 * m0 + 2 * hi];
        #pragma unroll
        for (int d = 0; d < 4; ++d) {
            v2f vb[8];
            #pragma unroll
            for (int m0 = 0; m0 < 8; ++m0) {
                const int mk = 4 * m0 + 2 * hi;
                vb[m0].x = cV[mk * VS + 16 * d + lo];
                vb[m0].y = cV[(mk + 1) * VS + 16 * d + lo];
            }
            #pragma unroll
            for (int m0 = 0; m0 < 8; ++m0)
                acc[d] = __builtin_amdgcn_wmma_f32_16x16x4_f32(
                            false, pa[m0], false, vb[m0], (short)0, acc[d], false, false);
        }
        cur ^= 1;
    }

    // ---- Epilogue: normalize, add residual, store [B][C][N] ----
    float inv[8];
    #pragma unroll
    for (int v = 0; v < 8; ++v) inv[v] = 1.0f / lrow[v];
    const float* xb = x   + (size_t)b * CH * NN;
    float*       ob = out + (size_t)b * CH * NN;
    #pragma unroll
    for (int d = 0; d < 4; ++d)
        #pragma unroll
        for (int v = 0; v < 8; ++v) {
            const int n = nbase + v + 8 * hi;
            const int c = 16 * d + lo;
            ob[(size_t)c * NN + n] = acc[d][v] * inv[v] + xb[(size_t)c * NN + n];
        }
}

extern "C" void kernel_launch(void* const* d_in, const int* in_sizes, int n_in,
                              void* d_out, int out_size, void* d_ws, size_t ws_size,
                              hipStream_t stream) {
    const float* x  = (const float*)d_in[0];
    const float* w1 = (const float*)d_in[1];
    const float* b1 = (const float*)d_in[2];
    const float* w2 = (const float*)d_in[3];
    const float* b2 = (const float*)d_in[4];
    const float* w3 = (const float*)d_in[5];
    const float* b3 = (const float*)d_in[6];
    float* out = (float*)d_out;

    float* Q = (float*)d_ws;                       // 12 MiB workspace total
    float* K = Q + (size_t)NB * NN * CH;
    float* V = K + (size_t)NB * NN * CH;

    qkv_kernel<<<128, 256, 0, stream>>>(x, w1, b1, w2, b2, w3, b3, Q, K, V);
    attn_kernel<<<NB * (NN / 64), 128, 0, stream>>>(Q, K, V, x, out);
}